// Decoder_67705864454690
// MI455X (gfx1250) — compile-verified
//
#include <hip/hip_runtime.h>
#include <hip/hip_bf16.h>

#define PI_F 3.14159265358979323846f

typedef __attribute__((ext_vector_type(2))) float v2f;
typedef __attribute__((ext_vector_type(8))) float v8f;

struct BandPtrs {
  const float* cw[6];
  const float* cb[6];
};

__device__ __forceinline__ float lrelu_f(float v){ return v > 0.f ? v : 0.2f * v; }

// ---------------------------------------------------------------------------
// CDNA5 f32 WMMA 16x16x4: D = A(16x4) * B(4x16) + C
// A frag: lane<16 -> M=lane, comps K={0,1}; lane>=16 -> M=lane-16, K={2,3}
// B frag: lane<16 -> N=lane, comps K={0,1}; lane>=16 -> N=lane-16, K={2,3}
// C/D:    VGPR r -> M = r + 8*(lane/16), N = lane%16
// ---------------------------------------------------------------------------
__device__ __forceinline__ v8f wmma4(v2f a, v2f b, v8f c){
#if __has_builtin(__builtin_amdgcn_wmma_f32_16x16x4_f32)
  return __builtin_amdgcn_wmma_f32_16x16x4_f32(false, a, false, b, (short)0, c, false, false);
#else
  // layout-consistent shuffle emulation (keeps kernel correct if builtin absent)
  const int lane  = threadIdx.x & 31;
  const int lhalf = lane >> 4;
  const int lrow  = lane & 15;
  for (int r = 0; r < 8; ++r){
    int M = r + 8 * lhalf;
    float acc = c[r];
    for (int k = 0; k < 4; ++k){
      float av = __shfl((k & 1) ? a.y : a.x, (M & 15) + ((k >> 1) << 4), 32);
      float bv = __shfl((k & 1) ? b.y : b.x, lrow + ((k >> 1) << 4), 32);
      acc += av * bv;
    }
    c[r] = acc;
  }
  return c;
#endif
}

// One 16x16 output tile of W(rows x 64) @ Z(64 x 16), Z resident in LDS [64][16].
__device__ __forceinline__ v8f gemm_tile(const float* __restrict__ W, int mbase, int rowLimit,
                                         const float* Zl, int lrow, int lhalf){
  v8f c = {0.f,0.f,0.f,0.f,0.f,0.f,0.f,0.f};
  int arow = mbase + lrow;
  if (arow > rowLimit - 1) arow = rowLimit - 1;   // clamp OOB weight rows (results masked at store)
  const float* wr = W + arow * 64;
  #pragma unroll
  for (int kk = 0; kk < 16; ++kk){
    const int k0 = kk * 4 + 2 * lhalf;
    v2f a, b;
    a.x = wr[k0];
    a.y = wr[k0 + 1];
    b.x = Zl[k0 * 16 + lrow];
    b.y = Zl[(k0 + 1) * 16 + lrow];
    c = wmma4(a, b, c);
  }
  return c;
}

// ---------------------------------------------------------------------------
// Stage 1: h0 = x @ up_lin_w + up_lin_b   -> (32, 64, 4)
// ---------------------------------------------------------------------------
__global__ void __launch_bounds__(256) uplin_kernel(const float* __restrict__ x,
    const float* __restrict__ w, const float* __restrict__ bias, float* __restrict__ outp){
  int idx = blockIdx.x * blockDim.x + threadIdx.x;
  if (idx >= 32 * 256) return;
  int oc = idx & 255, b = idx >> 8;
  float acc = bias[oc];
  for (int i = 0; i < 64; ++i) acc += x[b * 64 + i] * w[i * 256 + oc];
  outp[idx] = acc;
}

// ---------------------------------------------------------------------------
// fft_upsample(x, 2): exact closed form of irfft(pad(rfft(x)))*2.
// y_j = (1/n) * sum_i x_i * [1 + 2*sum_{k=1..n/2} cos(pi*k*(j-2i)/n)]
// ---------------------------------------------------------------------------
__global__ void __launch_bounds__(256) fftup_kernel(const float* __restrict__ in,
    float* __restrict__ outp, int n){
  int idx = blockIdx.x * blockDim.x + threadIdx.x;
  int N2 = 2 * n;
  if (idx >= 2048 * N2) return;
  int j = idx % N2, bc = idx / N2;
  const float* xr = in + bc * n;
  const float base = PI_F / (float)n;
  float acc = 0.f;
  for (int i = 0; i < n; ++i){
    float d = base * (float)(j - 2 * i);
    float m = 1.f;
    for (int k = 1; k <= n / 2; ++k) m += 2.f * cosf(d * (float)k);
    acc += xr[i] * m;
  }
  outp[idx] = acc / (float)n;
}

// ---------------------------------------------------------------------------
// conv3 (NCH, OIH, pad=1), optional leaky relu
// ---------------------------------------------------------------------------
__global__ void __launch_bounds__(256) conv3_kernel(const float* __restrict__ in,
    const float* __restrict__ w, const float* __restrict__ bias,
    float* __restrict__ outp, int T, int doLrelu){
  int idx = blockIdx.x * blockDim.x + threadIdx.x;
  if (idx >= 32 * 64 * T) return;
  int t = idx % T;
  int o = (idx / T) & 63;
  int b = idx / (64 * T);
  float acc = bias[o];
  const float* xin = in + b * 64 * T;
  const float* wr  = w + o * 192;
  for (int i = 0; i < 64; ++i){
    const float* xi = xin + i * T;
    if (t > 0)     acc += wr[i * 3 + 0] * xi[t - 1];
                   acc += wr[i * 3 + 1] * xi[t];
    if (t < T - 1) acc += wr[i * 3 + 2] * xi[t + 1];
  }
  if (doLrelu) acc = lrelu_f(acc);
  outp[idx] = acc;
}

// ---------------------------------------------------------------------------
// Per-band fused stack. grid = (64 column tiles over B*T=1024, 6 bands),
// block = 128 threads = 4 waves; each wave owns one 16-row tile.
// Produces amp/freq frames and per-frame noise filter coefficients.
// ---------------------------------------------------------------------------
__global__ void __launch_bounds__(128) band_stack_kernel(const float* __restrict__ h,
    const float* __restrict__ t_w,  const float* __restrict__ t_b,
    const float* __restrict__ bf_w, const float* __restrict__ bf_b,
    const float* __restrict__ oa_w, const float* __restrict__ oa_b,
    const float* __restrict__ of_w, const float* __restrict__ of_b,
    const float* __restrict__ nu_w, const float* __restrict__ nu_b,
    BandPtrs bp,
    float* __restrict__ ampb, float* __restrict__ frqb, float* __restrict__ coefb)
{
  __shared__ float Zp [64 * 16];
  __shared__ float Z2p[64 * 16];
  __shared__ float Up [64 * 16];

  const int band  = blockIdx.y;
  const int ntile = blockIdx.x;
  const int tid   = threadIdx.x;
  const int lane  = tid & 31;
  const int wave  = tid >> 5;
  const int lrow  = lane & 15;
  const int lhalf = lane >> 4;
  const int ncol  = ntile * 16 + lrow;   // global column (b*32 + t)
  const int cb    = ncol >> 5;
  const int ct    = ncol & 31;
  const int mbase = wave * 16;

  for (int idx = tid; idx < 1024; idx += 128){
    int row = idx >> 4, col = idx & 15;
    int n = ntile * 16 + col;
    Zp[row * 16 + col] = h[(n >> 5) * 2048 + row * 32 + (n & 31)];
  }
  __syncthreads();

  // 4 residual 1x1-conv layers: z = lrelu(W z + b + z)
  for (int j = 0; j < 4; ++j){
    const float* W  = t_w + (band * 4 + j) * 4096;
    const float* bb = t_b + (band * 4 + j) * 64;
    v8f c = gemm_tile(W, mbase, 64, Zp, lrow, lhalf);
    float nv[8];
    #pragma unroll
    for (int r = 0; r < 8; ++r){
      int row = mbase + r + 8 * lhalf;
      nv[r] = lrelu_f(c[r] + bb[row] + Zp[row * 16 + lrow]);
    }
    __syncthreads();
    #pragma unroll
    for (int r = 0; r < 8; ++r){
      int row = mbase + r + 8 * lhalf;
      Zp[row * 16 + lrow] = nv[r];
    }
    __syncthreads();
  }

  // band_final 1x1 conv -> Z2p
  {
    v8f c = gemm_tile(bf_w + band * 4096, mbase, 64, Zp, lrow, lhalf);
    const float* bb = bf_b + band * 64;
    #pragma unroll
    for (int r = 0; r < 8; ++r){
      int row = mbase + r + 8 * lhalf;
      Z2p[row * 16 + lrow] = c[r] + bb[row];
    }
    __syncthreads();
  }

  // amp / freq heads (32 osc rows each): waves 0-1 -> amp tiles, 2-3 -> freq tiles
  {
    const int isF = wave >> 1;
    const int mt  = wave & 1;
    const float* W  = (isF ? of_w : oa_w) + band * 2048;
    const float* bb = (isF ? of_b : oa_b) + band * 32;
    const float  lf = (band == 0) ? 0.05f : 0.01f;
    v8f c = gemm_tile(W, mt * 16, 32, Z2p, lrow, lhalf);
    float* dst = isF ? frqb : ampb;
    #pragma unroll
    for (int r = 0; r < 8; ++r){
      int row = mt * 16 + r + 8 * lhalf;           // oscillator index
      float v = c[r] + bb[row];
      if (isF) v = lf + (1.f - lf) / (1.f + expf(-v));
      else     v = fabsf(v);
      dst[band * 32768 + cb * 1024 + row * 32 + ct] = v;
    }
  }

  // noise_up: u = lrelu(W z2 + b)   (column-dedup of repeat(z,2))
  {
    v8f c = gemm_tile(nu_w + band * 4096, mbase, 64, Z2p, lrow, lhalf);
    const float* bb = nu_b + band * 64;
    #pragma unroll
    for (int r = 0; r < 8; ++r){
      int row = mbase + r + 8 * lhalf;
      Up[row * 16 + lrow] = lrelu_f(c[r] + bb[row]);
    }
    __syncthreads();
  }

  // noise filter coefficients: (nc x 64) @ u
  {
    const int NCv[6] = {5, 9, 17, 33, 65, 129};
    const int nc = NCv[band];
    const int ntiles = (nc + 15) >> 4;
    for (int mt = wave; mt < ntiles; mt += 4){
      v8f c = gemm_tile(bp.cw[band], mt * 16, nc, Up, lrow, lhalf);
      #pragma unroll
      for (int r = 0; r < 8; ++r){
        int row = mt * 16 + r + 8 * lhalf;
        if (row < nc)
          coefb[band * 147456 + row * 1024 + ncol] = c[r] + bp.cb[band][row];
      }
    }
  }
}

// ---------------------------------------------------------------------------
// Harmonic bank: linear-upsample amp/freq, cumsum(freq*pi), sum_osc sin*amp.
// One block per (batch); block-wide scan for the phase prefix.
// ---------------------------------------------------------------------------
__device__ __forceinline__ float interp32(const float* fr, float pos){
  int lo = (int)floorf(pos);
  lo = lo < 0 ? 0 : (lo > 31 ? 31 : lo);
  float w = pos - (float)lo;
  w = w < 0.f ? 0.f : (w > 1.f ? 1.f : w);
  int hi = lo + 1 > 31 ? 31 : lo + 1;
  return fr[lo] * (1.f - w) + fr[hi] * w;
}

__global__ void __launch_bounds__(256) harm_kernel(const float* __restrict__ ampb,
    const float* __restrict__ frqb, float* __restrict__ outp,
    int band, int bs, int boff, float inv_adj)
{
  __shared__ float part[256];
  __shared__ float fa[32];
  __shared__ float ff[32];
  const int b   = blockIdx.x;
  const int tid = threadIdx.x;
  const int L   = bs >> 8;
  const float scale = 32.f / (float)bs;
  float* orow = outp + b * 64512 + boff;
  const float* abase = ampb + band * 32768 + b * 1024;
  const float* fbase = frqb + band * 32768 + b * 1024;

  for (int osc = 0; osc < 32; ++osc){
    if (tid < 32){ fa[tid] = abase[osc * 32 + tid]; ff[tid] = fbase[osc * 32 + tid]; }
    __syncthreads();
    const int start = tid * L;
    float s = 0.f;
    for (int j = 0; j < L; ++j){
      float pos = ((float)(start + j) + 0.5f) * scale - 0.5f;
      s += interp32(ff, pos) * PI_F;
    }
    part[tid] = s;
    __syncthreads();
    for (int st = 1; st < 256; st <<= 1){
      float v = (tid >= st) ? part[tid - st] : 0.f;
      __syncthreads();
      part[tid] += v;
      __syncthreads();
    }
    float run = part[tid] - s;    // exclusive prefix of phase
    for (int j = 0; j < L; ++j){
      int t = start + j;
      float pos = ((float)t + 0.5f) * scale - 0.5f;
      run += interp32(ff, pos) * PI_F;
      float v = sinf(run) * interp32(fa, pos) * inv_adj;
      if (osc == 0) orow[t] = v; else orow[t] += v;
    }
    __syncthreads();
  }
}

// ---------------------------------------------------------------------------
// Filtered noise: per (batch, frame) direct rDFT of the noise excitation,
// scale by real coeffs, inverse rDFT, /ADJUST. Block per (frame, batch).
// ---------------------------------------------------------------------------
__global__ void __launch_bounds__(128) noise_kernel(const float* __restrict__ nz,
    const float* __restrict__ coefb, float* __restrict__ outp,
    int band, int spf, int nc, int boff, int bs, float inv_adj)
{
  __shared__ float xs[256];
  __shared__ float cr[132];
  __shared__ float ci[132];
  const int frame = blockIdx.x;
  const int b     = blockIdx.y;
  const int tid   = threadIdx.x;
  const float* src = nz + (b * 64 + frame) * spf;
  for (int s = tid; s < spf; s += 128) xs[s] = src[s];
  __syncthreads();
  const float w0 = 2.f * PI_F / (float)spf;
  for (int k = tid; k < nc; k += 128){
    float c = coefb[band * 147456 + k * 1024 + b * 32 + (frame >> 1)];
    if (band == 0 && k > 0) c = 0.f;            // low band keeps DC coeff only
    float nr = 0.f, ni = 0.f;
    for (int s = 0; s < spf; ++s){
      float sv, cv;
      sincosf(w0 * (float)(k * s), &sv, &cv);
      nr += xs[s] * cv;
      ni -= xs[s] * sv;
    }
    cr[k] = c * nr;
    ci[k] = c * ni;
  }
  __syncthreads();
  const float inv = inv_adj / (float)spf;
  float* orow = outp + b * 64512 + boff + bs + frame * spf;
  for (int s = tid; s < spf; s += 128){
    float y = cr[0];
    for (int k = 1; k < nc - 1; ++k){
      float sv, cv;
      sincosf(w0 * (float)(k * s), &sv, &cv);
      y += 2.f * (cr[k] * cv - ci[k] * sv);
    }
    y += cr[nc - 1] * ((s & 1) ? -1.f : 1.f);   // Nyquist bin (counted once)
    orow[s] = y * inv;
  }
}

// ---------------------------------------------------------------------------
extern "C" void kernel_launch(void* const* d_in, const int* in_sizes, int n_in,
                              void* d_out, int out_size, void* d_ws, size_t ws_size,
                              hipStream_t stream)
{
  const float* x         = (const float*)d_in[0];
  const float* up_lin_w  = (const float*)d_in[1];
  const float* up_lin_b  = (const float*)d_in[2];
  const float* up_conv_w = (const float*)d_in[3];
  const float* up_conv_b = (const float*)d_in[4];
  const float* up_fin_w  = (const float*)d_in[5];
  const float* up_fin_b  = (const float*)d_in[6];
  const float* t_w       = (const float*)d_in[7];
  const float* t_b       = (const float*)d_in[8];
  const float* bf_w      = (const float*)d_in[9];
  const float* bf_b      = (const float*)d_in[10];
  const float* oa_w      = (const float*)d_in[11];
  const float* oa_b      = (const float*)d_in[12];
  const float* of_w      = (const float*)d_in[13];
  const float* of_b      = (const float*)d_in[14];
  const float* nu_w      = (const float*)d_in[15];
  const float* nu_b      = (const float*)d_in[16];

  float* ws    = (float*)d_ws;
  float* bufA  = ws;                 // 65536
  float* bufB  = ws + 65536;         // 65536
  float* ampb  = ws + 131072;        // 6 * 32768
  float* frqb  = ws + 327680;        // 6 * 32768
  float* coefb = ws + 524288;        // 6 * 147456
  float* outp  = (float*)d_out;

  // Upsampling trunk: h0 -> (up x2, conv3+lrelu) x3 -> final conv3
  uplin_kernel<<<32, 256, 0, stream>>>(x, up_lin_w, up_lin_b, bufA);
  int n = 4;
  for (int i = 0; i < 3; ++i){
    int N2 = 2 * n;
    fftup_kernel<<<(2048 * N2 + 255) / 256, 256, 0, stream>>>(bufA, bufB, n);
    conv3_kernel<<<(32 * 64 * N2 + 255) / 256, 256, 0, stream>>>(
        bufB, up_conv_w + i * 12288, up_conv_b + i * 64, bufA, N2, 1);
    n = N2;
  }
  conv3_kernel<<<(32 * 64 * 32 + 255) / 256, 256, 0, stream>>>(
      bufA, up_fin_w, up_fin_b, bufB, 32, 0);

  // Fused per-band WMMA stack
  BandPtrs bp;
  for (int k = 0; k < 6; ++k){
    bp.cw[k] = (const float*)d_in[17 + k];
    bp.cb[k] = (const float*)d_in[23 + k];
  }
  band_stack_kernel<<<dim3(64, 6), 128, 0, stream>>>(bufB, t_w, t_b, bf_w, bf_b,
      oa_w, oa_b, of_w, of_b, nu_w, nu_b, bp, ampb, frqb, coefb);

  static const int   BS[6]  = {512, 1024, 2048, 4096, 8192, 16384};
  static const int   OFF[6] = {0, 1024, 3072, 7168, 15360, 31744};
  static const int   NCv[6] = {5, 9, 17, 33, 65, 129};
  static const float ADJ[6] = {0.05f, 0.03f, 0.05f, 0.25f, 1.0f, 20.0f};

  for (int k = 0; k < 6; ++k)
    harm_kernel<<<32, 256, 0, stream>>>(ampb, frqb, outp, k, BS[k], OFF[k], 1.0f / ADJ[k]);
  for (int k = 0; k < 6; ++k)
    noise_kernel<<<dim3(64, 32), 128, 0, stream>>>((const float*)d_in[29 + k], coefb,
        outp, k, BS[k] / 64, NCv[k], OFF[k], BS[k], 1.0f / ADJ[k]);
}